// SAGEMolGCN_48962627175092
// MI455X (gfx1250) — compile-verified
//
#include <hip/hip_runtime.h>
#include <hip/hip_bf16.h>

typedef __attribute__((ext_vector_type(2))) float v2f;
typedef __attribute__((ext_vector_type(8))) float v8f;

#define N_NODES 100000
#define N_EDGES 1600000
#define DDIM    128
#define NDSZ    ((size_t)N_NODES * DDIM)

// ---------------- zero ----------------
__global__ void zero_kernel(float* __restrict__ p, size_t n) {
    size_t i = (size_t)blockIdx.x * blockDim.x + threadIdx.x;
    size_t stride = (size_t)gridDim.x * blockDim.x;
    for (; i < n; i += stride) p[i] = 0.0f;
}

// ---------------- degree count ----------------
__global__ void count_kernel(const int* __restrict__ dst, float* __restrict__ cnt, int nEdges) {
    int i = blockIdx.x * blockDim.x + threadIdx.x;
    int stride = gridDim.x * blockDim.x;
    for (; i < nEdges; i += stride)
        unsafeAtomicAdd(&cnt[dst[i]], 1.0f);
}

__global__ void invert_kernel(float* __restrict__ cnt, int n) {
    int i = blockIdx.x * blockDim.x + threadIdx.x;
    if (i < n) cnt[i] = 1.0f / fmaxf(cnt[i], 1.0f);
}

// ---------------- scatter-add: agg[dst] += x[src], one wave per edge ----------------
__global__ void scatter_kernel(const float* __restrict__ x, const int* __restrict__ src,
                               const int* __restrict__ dst, float* __restrict__ agg, int nEdges) {
    int warp  = (blockIdx.x * blockDim.x + threadIdx.x) >> 5;
    int lane  = threadIdx.x & 31;
    int nWarp = (gridDim.x * blockDim.x) >> 5;
    for (int e = warp; e < nEdges; e += nWarp) {
        int s = src[e];
        int d = dst[e];
        const float4 v = ((const float4*)(x + (size_t)s * DDIM))[lane];
        float* ad = agg + (size_t)d * DDIM + lane * 4;
        unsafeAtomicAdd(ad + 0, v.x);
        unsafeAtomicAdd(ad + 1, v.y);
        unsafeAtomicAdd(ad + 2, v.z);
        unsafeAtomicAdd(ad + 3, v.w);
    }
}

// ---------------- fused SAGE GEMM: out = relu?((agg*inv) @ Wl^T + bl + x @ Wr^T) ----------------
// grid = N/16 blocks (row strips), block = 256 threads = 8 waves; wave w -> 16-col tile w.
// f32 WMMA 16x16x4: A lane l<16 holds (M=l, K=k0..k0+1), lane l>=16 holds (M=l-16, K=k0+2..k0+3).
// B lane n<16 holds (K=k0..k0+1, N=n), lane n>=16 holds (K=k0+2..k0+3, N=n-16).
// C/D: VGPR m = row m (lanes 0-15) / row m+8 (lanes 16-31), lane = column.
__global__ void __launch_bounds__(256)
sage_gemm_kernel(const float* __restrict__ agg, const float* __restrict__ inv,
                 const float* __restrict__ xin, const float* __restrict__ Wl,
                 const float* __restrict__ bl, const float* __restrict__ Wr,
                 float* __restrict__ xout, int doRelu) {
    const int rowBase = blockIdx.x * 16;
    const int wave    = threadIdx.x >> 5;
    const int lane    = threadIdx.x & 31;
    const int half    = lane >> 4;
    const int lid     = lane & 15;
    const int colBase = wave * 16;

    const int arow = rowBase + lid;                 // A-matrix row handled by this lane
    const float ainv = inv[arow];

    const float* aP  = agg + (size_t)arow * DDIM + 2 * half;
    const float* xP  = xin + (size_t)arow * DDIM + 2 * half;
    const float* wlP = Wl + (size_t)(colBase + lid) * DDIM + 2 * half;  // B[k][n] = Wl[n][k]
    const float* wrP = Wr + (size_t)(colBase + lid) * DDIM + 2 * half;

    v8f c = {};
#pragma unroll 4
    for (int k = 0; k < DDIM; k += 4) {
        v2f a, b;
        // neighbor-mean term: (agg*inv) @ Wl^T
        float2 av = *(const float2*)(aP + k);
        a.x = av.x * ainv; a.y = av.y * ainv;
        float2 bv = *(const float2*)(wlP + k);
        b.x = bv.x; b.y = bv.y;
        c = __builtin_amdgcn_wmma_f32_16x16x4_f32(false, a, false, b, (short)0, c, false, false);
        // root term: x @ Wr^T
        float2 xv = *(const float2*)(xP + k);
        a.x = xv.x; a.y = xv.y;
        float2 wv = *(const float2*)(wrP + k);
        b.x = wv.x; b.y = wv.y;
        c = __builtin_amdgcn_wmma_f32_16x16x4_f32(false, a, false, b, (short)0, c, false, false);
    }

    const float bias = bl[colBase + lid];
#pragma unroll
    for (int m = 0; m < 8; ++m) {
        float v = c[m] + bias;
        if (doRelu) v = fmaxf(v, 0.0f);
        xout[(size_t)(rowBase + m + 8 * half) * DDIM + colBase + lid] = v;
    }
}

// ---------------- column-wise min reduction ----------------
#define MIN_BLOCKS 256
__global__ void min_stage1(const float* __restrict__ h, float* __restrict__ part, int nRows) {
    int d = threadIdx.x;  // 128 threads
    float m = __builtin_inff();
    for (int r = blockIdx.x; r < nRows; r += gridDim.x)
        m = fminf(m, h[(size_t)r * DDIM + d]);
    part[(size_t)blockIdx.x * DDIM + d] = m;
}

__global__ void min_stage2(const float* __restrict__ part, float* __restrict__ out, int nPart) {
    int d = threadIdx.x;
    float m = __builtin_inff();
    for (int b = 0; b < nPart; ++b)
        m = fminf(m, part[(size_t)b * DDIM + d]);
    out[d] = m;
}

// ---------------- launcher ----------------
extern "C" void kernel_launch(void* const* d_in, const int* in_sizes, int n_in,
                              void* d_out, int out_size, void* d_ws, size_t ws_size,
                              hipStream_t stream) {
    const float* x  = (const float*)d_in[0];
    const int*   ei = (const int*)d_in[1];
    const float* Wl = (const float*)d_in[2];
    const float* bl = (const float*)d_in[3];
    const float* Wr = (const float*)d_in[4];

    const int nEdges = in_sizes[1] / 2;
    const int* src = ei;
    const int* dst = ei + nEdges;

    float* ws   = (float*)d_ws;
    float* AGG  = ws;                       // [N, D] neighbor sums
    float* H1   = ws + NDSZ;                // [N, D] ping
    float* H2   = ws + 2 * NDSZ;            // [N, D] pong
    float* CNT  = ws + 3 * NDSZ;            // [N] counts, inverted in place
    float* PART = CNT + N_NODES;            // [MIN_BLOCKS, D] partial mins

    const int gemmGrid = N_NODES / 16;      // 6250, exact

    // degree counts (shared by all layers)
    zero_kernel<<<1024, 256, 0, stream>>>(CNT, (size_t)N_NODES);
    count_kernel<<<4096, 256, 0, stream>>>(dst, CNT, nEdges);
    invert_kernel<<<(N_NODES + 255) / 256, 256, 0, stream>>>(CNT, N_NODES);

    // layer 0: x -> H1 (relu)
    zero_kernel<<<4096, 256, 0, stream>>>(AGG, NDSZ);
    scatter_kernel<<<4096, 256, 0, stream>>>(x, src, dst, AGG, nEdges);
    sage_gemm_kernel<<<gemmGrid, 256, 0, stream>>>(AGG, CNT, x, Wl, bl, Wr, H1, 1);

    // layer 1: H1 -> H2 (relu)
    zero_kernel<<<4096, 256, 0, stream>>>(AGG, NDSZ);
    scatter_kernel<<<4096, 256, 0, stream>>>(H1, src, dst, AGG, nEdges);
    sage_gemm_kernel<<<gemmGrid, 256, 0, stream>>>(AGG, CNT, H1,
                                                   Wl + DDIM * DDIM, bl + DDIM,
                                                   Wr + DDIM * DDIM, H2, 1);

    // layer 2: H2 -> H1 (no relu)
    zero_kernel<<<4096, 256, 0, stream>>>(AGG, NDSZ);
    scatter_kernel<<<4096, 256, 0, stream>>>(H2, src, dst, AGG, nEdges);
    sage_gemm_kernel<<<gemmGrid, 256, 0, stream>>>(AGG, CNT, H2,
                                                   Wl + 2 * DDIM * DDIM, bl + 2 * DDIM,
                                                   Wr + 2 * DDIM * DDIM, H1, 0);

    // column-wise min over nodes
    min_stage1<<<MIN_BLOCKS, DDIM, 0, stream>>>(H1, PART, N_NODES);
    min_stage2<<<1, DDIM, 0, stream>>>(PART, (float*)d_out, MIN_BLOCKS);
}